// SelfAttention_88510686036660
// MI455X (gfx1250) — compile-verified
//
#include <hip/hip_runtime.h>
#include <hip/hip_bf16.h>

// ---------------- problem constants (from reference) ----------------
constexpr int B_    = 32;
constexpr int DIM_  = 4096;
constexpr int NQ_   = 32;
constexpr int NKV_  = 8;
constexpr int HD_   = 128;
constexpr int NREP_ = NQ_ / NKV_;   // 4
constexpr int MAXSEQ_ = 2048;

typedef float v2f __attribute__((ext_vector_type(2)));
typedef float v8f __attribute__((ext_vector_type(8)));

// =====================================================================
// GEMM: C(32 x N) = A(32 x K) * W(K x N), all fp32 row-major.
// One wave computes one 16-wide N tile for all 32 rows (two 16x16 C
// tiles) using V_WMMA_F32_16X16X4_F32, K advanced 4 per step.
// A-fragment layout (ISA 7.12.2, 32-bit A 16x4): lane<16 -> K=k+0,k+1,
// lane>=16 -> K=k+2,k+3; lane&15 selects M row.  B mirrored (lane = N).
// =====================================================================
__global__ __launch_bounds__(256)
void gemm_m32_wmma(const float* __restrict__ A,
                   const float* __restrict__ W,
                   float* __restrict__ C,
                   int K, int N)
{
    const int lane  = threadIdx.x & 31;
    const int wave  = threadIdx.x >> 5;
    const int ntile = blockIdx.x * (blockDim.x >> 5) + wave;
    if (ntile * 16 >= N) return;               // wave-uniform, EXEC stays full

    const int col   = ntile * 16 + (lane & 15);
    const int row   = lane & 15;
    const int khalf = (lane >> 4) * 2;         // 0 or 2

    v8f acc0 = {};                             // rows 0..15
    v8f acc1 = {};                             // rows 16..31

    const float* a0p = A + (size_t)row        * K + khalf;
    const float* a1p = A + (size_t)(row + 16) * K + khalf;
    const float* wp  = W + (size_t)khalf * N + col;

    for (int k = 0; k < K; k += 4) {
        v2f a0 = { a0p[0], a0p[1] };           // contiguous in K -> b64 load
        v2f a1 = { a1p[0], a1p[1] };
        v2f b  = { wp[0],  wp[N]  };           // B[k+khalf][col], B[k+khalf+1][col]
        acc0 = __builtin_amdgcn_wmma_f32_16x16x4_f32(
                   false, a0, false, b, (short)0, acc0, false, false);
        acc1 = __builtin_amdgcn_wmma_f32_16x16x4_f32(
                   false, a1, false, b, (short)0, acc1, false, false);
        a0p += 4; a1p += 4; wp += (size_t)4 * N;
    }

    // C/D layout: VGPR i, lanes 0-15 -> M=i, lanes 16-31 -> M=i+8
    #pragma unroll
    for (int i = 0; i < 8; ++i) {
        const int r = (lane < 16) ? i : (i + 8);
        C[(size_t)r        * N + col] = acc0[i];
        C[(size_t)(r + 16) * N + col] = acc1[i];
    }
}

// =====================================================================
// RoPE on staged q (B*NQ*HD) and k (B*NKV*HD); freqs are (HD/2) each.
// One thread per complex pair.
// =====================================================================
__global__ __launch_bounds__(256)
void rope_kernel(float* __restrict__ q, float* __restrict__ k,
                 const float* __restrict__ fr, const float* __restrict__ fi)
{
    const int idx = blockIdx.x * blockDim.x + threadIdx.x;
    const int NQP = B_ * NQ_ * HD_ / 2;        // 65536
    const int NKP = B_ * NKV_ * HD_ / 2;       // 16384

    if (idx < NQP) {
        const int p = idx & (HD_ / 2 - 1);
        const float r = fr[p], im = fi[p];
        const float x0 = q[2 * idx], x1 = q[2 * idx + 1];
        q[2 * idx]     = x0 * r  - x1 * im;
        q[2 * idx + 1] = x0 * im + x1 * r;
    }
    if (idx < NKP) {
        const int p = idx & (HD_ / 2 - 1);
        const float r = fr[p], im = fi[p];
        const float x0 = k[2 * idx], x1 = k[2 * idx + 1];
        k[2 * idx]     = x0 * r  - x1 * im;
        k[2 * idx + 1] = x0 * im + x1 * r;
    }
}

// =====================================================================
// GQA decode attention: one workgroup per (b, kv-head g).  256 threads.
// Phase 1: one WAVE per key row j: 32 lanes x float4 == the full 128-dim
//          K row (coalesced 512B), dot vs 4 Q heads, shfl_xor reduction.
// Phase 2: block softmax per rep (max + sum reductions in LDS).
// Phase 3: j-range split across thread halves so every V element is
//          loaded exactly once (coalesced); halves combined via LDS.
// =====================================================================
__global__ __launch_bounds__(256)
void attn_kernel(const float* __restrict__ xq,   // B x NQ x HD (roped)
                 const float* __restrict__ xk,   // B x NKV x HD (roped new row)
                 const float* __restrict__ xv,   // B x NKV x HD (new row)
                 const float* __restrict__ cache_k,
                 const float* __restrict__ cache_v,
                 const int*   __restrict__ start_pos_p,
                 float* __restrict__ out)        // B x NQ x HD
{
    const int b    = blockIdx.x / NKV_;
    const int g    = blockIdx.x % NKV_;
    const int tid  = threadIdx.x;
    const int lane = tid & 31;
    const int wv   = tid >> 5;                   // wave 0..7
    const int sp   = *start_pos_p;               // 2048

    __shared__ float sQ[NREP_ * HD_];            // 2 KB (reused as sP in phase 3)
    __shared__ float sS[NREP_ * MAXSEQ_];        // 32 KB
    __shared__ float red[256];

    for (int i = tid; i < NREP_ * HD_; i += 256) {
        const int r = i / HD_, d = i % HD_;
        sQ[i] = xq[((size_t)b * NQ_ + g * NREP_ + r) * HD_ + d];
    }
    __syncthreads();

    const float scale = 0.088388347648318447f;   // 1/sqrt(128)

    // ---- phase 1: scores, one wave per key row ----
    float4 qf[NREP_];
    #pragma unroll
    for (int r = 0; r < NREP_; ++r)
        qf[r] = *(const float4*)&sQ[r * HD_ + lane * 4];

    const float* knew = xk + ((size_t)b * NKV_ + g) * HD_;
    for (int j = wv; j < sp; j += 8) {
        const float* kp = (j == sp - 1)
            ? knew
            : (cache_k + (((size_t)b * MAXSEQ_ + j) * NKV_ + g) * HD_);
        const float4 kv = *(const float4*)(kp + lane * 4);   // coalesced b128
        float acc[NREP_];
        #pragma unroll
        for (int r = 0; r < NREP_; ++r)
            acc[r] = qf[r].x * kv.x + qf[r].y * kv.y + qf[r].z * kv.z + qf[r].w * kv.w;
        // wave32 tree reduction
        #pragma unroll
        for (int off = 16; off > 0; off >>= 1) {
            #pragma unroll
            for (int r = 0; r < NREP_; ++r)
                acc[r] += __shfl_xor(acc[r], off, 32);
        }
        if (lane == 0) {
            #pragma unroll
            for (int r = 0; r < NREP_; ++r)
                sS[r * MAXSEQ_ + j] = acc[r] * scale;
        }
    }
    __syncthreads();

    // ---- phase 2: softmax (per rep) ----
    float inv_sum[NREP_];
    for (int r = 0; r < NREP_; ++r) {
        float m = -3.402823466e38f;
        for (int j = tid; j < sp; j += 256) m = fmaxf(m, sS[r * MAXSEQ_ + j]);
        red[tid] = m; __syncthreads();
        for (int s = 128; s > 0; s >>= 1) {
            if (tid < s) red[tid] = fmaxf(red[tid], red[tid + s]);
            __syncthreads();
        }
        m = red[0]; __syncthreads();

        float sum = 0.f;
        for (int j = tid; j < sp; j += 256) {
            const float p = __expf(sS[r * MAXSEQ_ + j] - m);
            sS[r * MAXSEQ_ + j] = p;
            sum += p;
        }
        red[tid] = sum; __syncthreads();
        for (int s = 128; s > 0; s >>= 1) {
            if (tid < s) red[tid] += red[tid + s];
            __syncthreads();
        }
        inv_sum[r] = 1.0f / red[0]; __syncthreads();
    }

    // ---- phase 3: out = P @ V, j-range split so V is read once ----
    const int half = tid >> 7;                   // wave-uniform (waves 0-3 / 4-7)
    const int d    = tid & (HD_ - 1);
    const int jbeg = half ? (sp >> 1) : 0;
    const int jend = half ? (sp - 1)  : (sp >> 1);

    float acc[NREP_] = {0.f, 0.f, 0.f, 0.f};
    const float* vcol = cache_v + (((size_t)b * MAXSEQ_) * NKV_ + g) * HD_ + d;
    for (int j = jbeg; j < jend; ++j) {
        const float v = vcol[(size_t)j * (NKV_ * HD_)];      // coalesced across lanes
        #pragma unroll
        for (int r = 0; r < NREP_; ++r) acc[r] += sS[r * MAXSEQ_ + j] * v;
    }
    if (half) {                                  // staged new V row at sp-1
        const float v = xv[((size_t)b * NKV_ + g) * HD_ + d];
        #pragma unroll
        for (int r = 0; r < NREP_; ++r) acc[r] += sS[r * MAXSEQ_ + (sp - 1)] * v;
    }

    float* sP = sQ;                              // reuse 512-float LDS buffer
    if (!half) {
        #pragma unroll
        for (int r = 0; r < NREP_; ++r) sP[r * HD_ + d] = acc[r];
    }
    __syncthreads();
    if (half) {
        #pragma unroll
        for (int r = 0; r < NREP_; ++r) {
            const float o = (acc[r] + sP[r * HD_ + d]) * inv_sum[r];
            out[((size_t)b * NQ_ + g * NREP_ + r) * HD_ + d] = o;
        }
    }
}

// =====================================================================
extern "C" void kernel_launch(void* const* d_in, const int* in_sizes, int n_in,
                              void* d_out, int out_size, void* d_ws, size_t ws_size,
                              hipStream_t stream) {
    const float* x   = (const float*)d_in[0];
    const int*   sp  = (const int*)  d_in[1];
    const float* fr  = (const float*)d_in[2];
    const float* fi  = (const float*)d_in[3];
    const float* wq  = (const float*)d_in[4];
    const float* wk  = (const float*)d_in[5];
    const float* wv  = (const float*)d_in[6];
    const float* wo  = (const float*)d_in[7];
    const float* ck  = (const float*)d_in[8];
    const float* cv  = (const float*)d_in[9];
    float* out = (float*)d_out;

    float* ws  = (float*)d_ws;
    float* xq  = ws;                                   // 131072 floats
    float* xk  = xq  + (size_t)B_ * NQ_  * HD_;        //  32768
    float* xv  = xk  + (size_t)B_ * NKV_ * HD_;        //  32768
    float* att = xv  + (size_t)B_ * NKV_ * HD_;        // 131072  (~1.3 MB total)

    // QKV projections (WMMA f32 GEMMs). 8 waves/block, 1 N-tile/wave.
    gemm_m32_wmma<<<(NQ_  * HD_) / (16 * 8), 256, 0, stream>>>(x, wq, xq, DIM_, NQ_  * HD_);
    gemm_m32_wmma<<<(NKV_ * HD_) / (16 * 8), 256, 0, stream>>>(x, wk, xk, DIM_, NKV_ * HD_);
    gemm_m32_wmma<<<(NKV_ * HD_) / (16 * 8), 256, 0, stream>>>(x, wv, xv, DIM_, NKV_ * HD_);

    // RoPE on q and new k row.
    rope_kernel<<<(B_ * NQ_ * HD_ / 2) / 256, 256, 0, stream>>>(xq, xk, fr, fi);

    // Attention over the KV cache (one workgroup per (batch, kv-head)).
    attn_kernel<<<B_ * NKV_, 256, 0, stream>>>(xq, xk, xv, ck, cv, sp, att);

    // Output projection.
    gemm_m32_wmma<<<DIM_ / (16 * 8), 256, 0, stream>>>(att, wo, out, NQ_ * HD_, DIM_);
}